// VanillaEgnnLayer_23871428231773
// MI455X (gfx1250) — compile-verified
//
#include <hip/hip_runtime.h>
#include <hip/hip_bf16.h>

// EGNN layer, fused, fp32 WMMA (V_WMMA_F32_16X16X4_F32) on gfx1250.
// B=8, N=128, D_MODEL=256, D_EDGE=64.
// Round 3: fix async builtin pointer types (AS1 int* / AS3 int*, non-const).

typedef __attribute__((ext_vector_type(2))) float v2f;
typedef __attribute__((ext_vector_type(8))) float v8f;

#define D_MODEL 256
#define NNODE   128
#define NBATCH  8
#define NROWS   (NBATCH * NNODE)   // 1024

#ifndef __has_builtin
#define __has_builtin(x) 0
#endif

#if __has_builtin(__builtin_amdgcn_global_load_async_to_lds_b32) && \
    __has_builtin(__builtin_amdgcn_s_wait_asynccnt)
#define ASYNC_LDS 1
#else
#define ASYNC_LDS 0
#endif

#define AS1_INT(p) ((__attribute__((address_space(1))) int*)(p))
#define AS3_INT(p) ((__attribute__((address_space(3))) int*)(p))

__device__ __forceinline__ v8f wmma4(v2f a, v2f b, v8f c) {
  // D = A(16x4) * B(4x16) + C(16x16), fp32
  return __builtin_amdgcn_wmma_f32_16x16x4_f32(false, a, false, b, (short)0, c,
                                               false, false);
}

__device__ __forceinline__ float silu_f(float x) {
  return x / (1.f + __expf(-x));
}

// Stage npairs K-pairs x 16 cols of W[:, col0:col0+16] into LDS pair layout:
//   BpDst[(kk*16+n)*2 + s] = W[(2kk+s)*256 + col0+n]
// Async path: per-lane GLOBAL_LOAD_ASYNC_TO_LDS_B32 (ASYNCcnt); the pair
// interleave keeps the WMMA inner loop at one ds_load_b64 per wmma.
__device__ __forceinline__ void stage_pairs(const float* __restrict__ Wsrc,
                                            int col0, int npairs,
                                            float* BpDst) {
  for (int idx = threadIdx.x; idx < npairs * 16; idx += 256) {
    const int kk = idx >> 4, n = idx & 15;
    float* g0 = const_cast<float*>(Wsrc + (size_t)(2 * kk) * D_MODEL + col0 + n);
#if ASYNC_LDS
    __builtin_amdgcn_global_load_async_to_lds_b32(
        AS1_INT(g0), AS3_INT(BpDst + idx * 2), 0, 0);
    __builtin_amdgcn_global_load_async_to_lds_b32(
        AS1_INT(g0 + D_MODEL), AS3_INT(BpDst + idx * 2 + 1), 0, 0);
#else
    BpDst[idx * 2 + 0] = g0[0];
    BpDst[idx * 2 + 1] = g0[D_MODEL];
#endif
  }
}

template <int N>
__device__ __forceinline__ void async_wait() {
#if ASYNC_LDS
  __builtin_amdgcn_s_wait_asynccnt(N);
#endif
}

// ---------------------------------------------------------------------------
// Kernel 1: node projections.
//   HiE = inv @ W_e1[0:256]   + b_e1     (bias folded once here)
//   HjE = inv @ W_e1[256:512]
//   HiX = inv @ W_x1[0:256]   + b_x1
//   HjX = inv @ W_x1[256:512]
// One wave computes one 16x16 output tile. 4096 wave-tiles.
// ---------------------------------------------------------------------------
__global__ __launch_bounds__(256) void node_proj_kernel(
    const float* __restrict__ X,
    const float* __restrict__ We1, const float* __restrict__ be1,
    const float* __restrict__ Wx1, const float* __restrict__ bx1,
    float* __restrict__ HiE, float* __restrict__ HjE,
    float* __restrict__ HiX, float* __restrict__ HjX) {
  const int lane = threadIdx.x & 31;
  const int half = lane >> 4;
  const int ln   = lane & 15;
  const int wt   = blockIdx.x * 8 + (threadIdx.x >> 5);  // 0..4095
  const int mat  = wt >> 10;
  const int rem  = wt & 1023;
  const int mt   = rem >> 4;   // 0..63  (16-row tile of 1024 rows)
  const int nt   = rem & 15;   // 0..15  (16-col tile of 256 cols)
  const int col0 = nt * 16;

  const float* W;
  const float* bias;
  float* Out;
  if (mat == 0)      { W = We1;                 bias = be1;     Out = HiE; }
  else if (mat == 1) { W = We1 + 256 * D_MODEL; bias = nullptr; Out = HjE; }
  else if (mat == 2) { W = Wx1;                 bias = bx1;     Out = HiX; }
  else               { W = Wx1 + 256 * D_MODEL; bias = nullptr; Out = HjX; }

  v8f acc = {};
  if (bias) {
    float bv = bias[col0 + ln];
#pragma unroll
    for (int v = 0; v < 8; ++v) acc[v] = bv;
  }

  const float* arow = X + (size_t)(mt * 16 + ln) * D_MODEL + half * 2;
#pragma unroll 8
  for (int ks = 0; ks < 64; ++ks) {           // K = 256, 4 per step
    v2f a = *(const v2f*)(arow + 4 * ks);
    int k = 4 * ks + half * 2;
    v2f b;
    b.x = W[(size_t)k * D_MODEL + col0 + ln];
    b.y = W[(size_t)(k + 1) * D_MODEL + col0 + ln];
    acc = wmma4(a, b, acc);
  }
#pragma unroll
  for (int v = 0; v < 8; ++v)
    Out[(size_t)(mt * 16 + v + half * 8) * D_MODEL + col0 + ln] = acc[v];
}

// ---------------------------------------------------------------------------
// Kernel 2: fused edge pipeline. One workgroup per (b,i); 8 waves, each wave
// owns a 16-j edge tile. Per branch:
//   pre  = HiE[i] + HjE[j] + ef64 @ W1[512:576] + sq * W1[576]   (WMMA + rank1)
//   msg  = silu(pre)                          -> A-regs via LDS transpose
//   msg2 = silu(msg @ W2 + b2)                (WMMA, W2 async-staged thru LDS)
//   att  = (sigmoid)(msg2 . Wv + bv)          (shfl reduce)
//   e-branch: nmsg += msg2 * att * adj ; x-branch: coord weights
// ---------------------------------------------------------------------------
template <bool IS_E>
__device__ __forceinline__ void run_branch(
    int wg, int b, int j0, int half, int ln, const v2f* aE,
    const float* __restrict__ W1e, const float* __restrict__ W1last,
    const float* __restrict__ Hi, const float* __restrict__ Hj,
    const float* __restrict__ W2, const float* __restrict__ b2,
    const float* __restrict__ Wv, float bv,
    const float* __restrict__ adjrow,
    float* Bp0, float* Bp1, float* scrW,
    const float* sqb, const float* rnb,
    float* attb_w, float* nmsg) {
  v2f aA[16][4];       // msg in WMMA A layout, 128 VGPRs
  float sqm[8];
#pragma unroll
  for (int v = 0; v < 8; ++v) sqm[v] = sqb[j0 + half * 8 + v];

  // ----- layer 1: 16x64 ef tile times W1e (K=64) + assembly + silu -----
  stage_pairs(W1e, 0, 32, Bp0);                     // prefetch block 0
#pragma unroll
  for (int nt = 0; nt < 16; ++nt) {
    const int col0 = nt * 16;
    float* Bcur = (nt & 1) ? Bp1 : Bp0;
    if (nt < 15) {                                  // prefetch next block
      stage_pairs(W1e, col0 + 16, 32, (nt & 1) ? Bp0 : Bp1);
      async_wait<4>();                              // 4 in flight = next block
    } else {
      async_wait<0>();
    }
    __syncthreads();
    v8f acc = {};
#pragma unroll
    for (int t = 0; t < 16; ++t) {
      v2f bb = *(const v2f*)&Bcur[((2 * t + half) * 16 + ln) * 2];
      acc = wmma4(aE[t], bb, acc);
    }
    float hi_n = Hi[(size_t)wg * D_MODEL + col0 + ln];
    float w576 = W1last[col0 + ln];
#pragma unroll
    for (int v = 0; v < 8; ++v) {
      int m = v + half * 8;
      float pre = acc[v] + hi_n +
                  Hj[(size_t)(b * NNODE + j0 + m) * D_MODEL + col0 + ln] +
                  sqm[v] * w576;
      scrW[m * 18 + ln] = silu_f(pre);   // D-layout -> padded LDS
    }
    __builtin_amdgcn_wave_barrier();
#pragma unroll
    for (int t = 0; t < 4; ++t)          // read back in A layout
      aA[nt][t] = *(const v2f*)&scrW[ln * 18 + 4 * t + half * 2];
    __builtin_amdgcn_wave_barrier();
    __syncthreads();                     // buffer free before reuse
  }

  // ----- layer 2: msg(16x256) @ W2(256x256), K async-staged via LDS -----
  float r2[16][8];
  float attp[8];
#pragma unroll
  for (int v = 0; v < 8; ++v) attp[v] = 0.f;

  stage_pairs(W2, 0, 128, Bp0);                     // prefetch block 0
#pragma unroll
  for (int nt = 0; nt < 16; ++nt) {
    const int col0 = nt * 16;
    float* Bcur = (nt & 1) ? Bp1 : Bp0;
    if (nt < 15) {
      stage_pairs(W2, col0 + 16, 128, (nt & 1) ? Bp0 : Bp1);
      async_wait<16>();                             // 16 in flight = next block
    } else {
      async_wait<0>();
    }
    __syncthreads();
    v8f acc;
    float bias_n = b2[col0 + ln];
#pragma unroll
    for (int v = 0; v < 8; ++v) acc[v] = bias_n;
#pragma unroll
    for (int ks = 0; ks < 64; ++ks) {
      v2f bb = *(const v2f*)&Bcur[((2 * ks + half) * 16 + ln) * 2];
      acc = wmma4(aA[ks >> 2][ks & 3], bb, acc);
    }
    float wv = Wv[col0 + ln];
#pragma unroll
    for (int v = 0; v < 8; ++v) {
      float s = silu_f(acc[v]);
      r2[nt][v] = s;
      attp[v] += s * wv;                 // per-lane partial of msg2 . Wv
    }
    __syncthreads();                     // buffer free before reuse
  }

  // ----- attention: reduce over n (16 lanes of each half) -----
#pragma unroll
  for (int v = 0; v < 8; ++v) {
    float t = attp[v];
    t += __shfl_xor(t, 1, 32);
    t += __shfl_xor(t, 2, 32);
    t += __shfl_xor(t, 4, 32);
    t += __shfl_xor(t, 8, 32);
    attp[v] = t;
  }
  if (ln == 0) {                          // lanes 0 (rows 0-7) and 16 (rows 8-15)
#pragma unroll
    for (int v = 0; v < 8; ++v) {
      int m = v + half * 8;
      int j = j0 + m;
      float a = attp[v] + bv;
      if (IS_E) a = 1.f / (1.f + __expf(-a));  // sigmoid (e-branch only)
      a *= adjrow[j];
      if (!IS_E) a *= rnb[j];                  // fold 1/normaliser (x-branch)
      attb_w[m] = a;
    }
  }
  __builtin_amdgcn_wave_barrier();

  if (IS_E) {                             // node_msg += msg2 * att * adj
    float aw[8];
#pragma unroll
    for (int v = 0; v < 8; ++v) aw[v] = attb_w[v + half * 8];
#pragma unroll
    for (int nt = 0; nt < 16; ++nt) {
      float s = 0.f;
#pragma unroll
      for (int v = 0; v < 8; ++v) s += r2[nt][v] * aw[v];
      s += __shfl_xor(s, 16, 32);         // other half's rows
      if (half == 0) atomicAdd(&nmsg[nt * 16 + ln], s);
    }
  }
}

__global__ __launch_bounds__(256, 1) void egnn_edge_kernel(
    const float* __restrict__ coords, const float* __restrict__ adjm,
    const float* __restrict__ mask, const float* __restrict__ ef,
    const float* __restrict__ We1, const float* __restrict__ Watt,
    const float* __restrict__ batt, const float* __restrict__ We2,
    const float* __restrict__ be2, const float* __restrict__ Wx1,
    const float* __restrict__ Wx3, const float* __restrict__ bx3,
    const float* __restrict__ Wx2, const float* __restrict__ bx2,
    const float* __restrict__ HiE, const float* __restrict__ HjE,
    const float* __restrict__ HiX, const float* __restrict__ HjX,
    float* __restrict__ nmsg_out, float* __restrict__ out_coords) {
  __shared__ __align__(16) float Bp[2][4096];       // 2x16KB async dbl buffer
  __shared__ __align__(16) float scr[8][16 * 18];   // per-wave transpose pad
  __shared__ float sqb[NNODE], rnb[NNODE];
  __shared__ float cjb[NNODE][3];
  __shared__ float attb[8][16];
  __shared__ float nmsg[D_MODEL];
  __shared__ float cacc[3];
  __shared__ float nnacc;

  const int tid  = threadIdx.x;
  const int wave = tid >> 5;
  const int lane = tid & 31;
  const int half = lane >> 4;
  const int ln   = lane & 15;
  const int wg   = blockIdx.x;      // b*128 + i
  const int b    = wg >> 7;
  const int j0   = wave * 16;

  if (tid < D_MODEL) nmsg[tid] = 0.f;
  if (tid == 0) { cacc[0] = cacc[1] = cacc[2] = 0.f; nnacc = 1.f; }
  __syncthreads();

  const float cix = coords[(size_t)wg * 3 + 0];
  const float ciy = coords[(size_t)wg * 3 + 1];
  const float ciz = coords[(size_t)wg * 3 + 2];
  if (tid < NNODE) {
    int j = tid;
    float cjx = coords[(size_t)(b * NNODE + j) * 3 + 0];
    float cjy = coords[(size_t)(b * NNODE + j) * 3 + 1];
    float cjz = coords[(size_t)(b * NNODE + j) * 3 + 2];
    float dx = cix - cjx, dy = ciy - cjy, dz = ciz - cjz;
    float sq = dx * dx + dy * dy + dz * dz;
    sqb[j] = sq;
    rnb[j] = 1.f / (sqrtf(sq + 1e-5f) + 1.f);
    cjb[j][0] = cjx; cjb[j][1] = cjy; cjb[j][2] = cjz;
    atomicAdd(&nnacc, mask[b * NNODE + j]);
  }
  __syncthreads();

  // edge-feature A tile (16j x 64), register resident, reused by both branches
  v2f aE[16];
  {
    const float* efb = ef + ((size_t)wg * NNODE + j0 + ln) * 64 + half * 2;
#pragma unroll
    for (int t = 0; t < 16; ++t) aE[t] = *(const v2f*)(efb + 4 * t);
  }
  const float* adjrow = adjm + (size_t)wg * NNODE;

  run_branch<true>(wg, b, j0, half, ln, aE,
                   We1 + 512 * D_MODEL, We1 + 576 * D_MODEL,
                   HiE, HjE, We2, be2, Watt, batt[0], adjrow,
                   Bp[0], Bp[1], &scr[wave][0], sqb, rnb,
                   &attb[wave][0], nmsg);

  run_branch<false>(wg, b, j0, half, ln, aE,
                    Wx1 + 512 * D_MODEL, Wx1 + 576 * D_MODEL,
                    HiX, HjX, Wx2, bx2, Wx3, bx3[0], adjrow,
                    Bp[0], Bp[1], &scr[wave][0], sqb, rnb,
                    &attb[wave][0], nmsg);

  // coord updates: sum_j diff * edge_attn * adj / norm
  if (lane < 16) {
    int j = j0 + lane;
    float w = attb[wave][lane];
    atomicAdd(&cacc[0], (cix - cjb[j][0]) * w);
    atomicAdd(&cacc[1], (ciy - cjb[j][1]) * w);
    atomicAdd(&cacc[2], (ciz - cjb[j][2]) * w);
  }
  __syncthreads();

  if (tid < D_MODEL) nmsg_out[(size_t)wg * D_MODEL + tid] = nmsg[tid];
  if (tid < 3) {
    float upd = cacc[tid] / nnacc;
    out_coords[(size_t)wg * 3 + tid] =
        (coords[(size_t)wg * 3 + tid] + upd) * mask[wg];
  }
}

// ---------------------------------------------------------------------------
// Kernel 3: h1 = silu(concat(inv, nmsg) @ W_h1 + b_h1)   [1024,512]@[512,256]
// ---------------------------------------------------------------------------
__global__ __launch_bounds__(256) void h1_kernel(
    const float* __restrict__ X, const float* __restrict__ NM,
    const float* __restrict__ Wh1, const float* __restrict__ bh1,
    float* __restrict__ H1) {
  const int lane = threadIdx.x & 31;
  const int half = lane >> 4;
  const int ln   = lane & 15;
  const int wt   = blockIdx.x * 8 + (threadIdx.x >> 5);  // 0..1023
  const int mt   = wt >> 4;
  const int nt   = wt & 15;
  const int col0 = nt * 16;

  v8f acc = {};
  float bv = bh1[col0 + ln];
#pragma unroll
  for (int v = 0; v < 8; ++v) acc[v] = bv;

  const float* arow0 = X  + (size_t)(mt * 16 + ln) * D_MODEL + half * 2;
  const float* arow1 = NM + (size_t)(mt * 16 + ln) * D_MODEL + half * 2;
#pragma unroll 8
  for (int ks = 0; ks < 128; ++ks) {      // K = 512
    v2f a = (ks < 64) ? *(const v2f*)(arow0 + 4 * ks)
                      : *(const v2f*)(arow1 + 4 * (ks - 64));
    int k = 4 * ks + half * 2;
    v2f bvec;
    bvec.x = Wh1[(size_t)k * D_MODEL + col0 + ln];
    bvec.y = Wh1[(size_t)(k + 1) * D_MODEL + col0 + ln];
    acc = wmma4(a, bvec, acc);
  }
#pragma unroll
  for (int v = 0; v < 8; ++v)
    H1[(size_t)(mt * 16 + v + half * 8) * D_MODEL + col0 + ln] = silu_f(acc[v]);
}

// ---------------------------------------------------------------------------
// Kernel 4: out_feats = (h1 @ W_h2 + b_h2) * mask
// ---------------------------------------------------------------------------
__global__ __launch_bounds__(256) void h2_kernel(
    const float* __restrict__ H1, const float* __restrict__ Wh2,
    const float* __restrict__ bh2, const float* __restrict__ mask,
    float* __restrict__ out_feats) {
  const int lane = threadIdx.x & 31;
  const int half = lane >> 4;
  const int ln   = lane & 15;
  const int wt   = blockIdx.x * 8 + (threadIdx.x >> 5);
  const int mt   = wt >> 4;
  const int nt   = wt & 15;
  const int col0 = nt * 16;

  v8f acc = {};
  float bv = bh2[col0 + ln];
#pragma unroll
  for (int v = 0; v < 8; ++v) acc[v] = bv;

  const float* arow = H1 + (size_t)(mt * 16 + ln) * D_MODEL + half * 2;
#pragma unroll 8
  for (int ks = 0; ks < 64; ++ks) {
    v2f a = *(const v2f*)(arow + 4 * ks);
    int k = 4 * ks + half * 2;
    v2f bvec;
    bvec.x = Wh2[(size_t)k * D_MODEL + col0 + ln];
    bvec.y = Wh2[(size_t)(k + 1) * D_MODEL + col0 + ln];
    acc = wmma4(a, bvec, acc);
  }
#pragma unroll
  for (int v = 0; v < 8; ++v) {
    int row = mt * 16 + v + half * 8;
    out_feats[(size_t)row * D_MODEL + col0 + ln] = acc[v] * mask[row];
  }
}

// ---------------------------------------------------------------------------
extern "C" void kernel_launch(void* const* d_in, const int* in_sizes, int n_in,
                              void* d_out, int out_size, void* d_ws,
                              size_t ws_size, hipStream_t stream) {
  (void)in_sizes; (void)n_in; (void)out_size; (void)ws_size;

  const float* coords = (const float*)d_in[0];
  const float* inv    = (const float*)d_in[1];
  const float* adj    = (const float*)d_in[2];
  const float* mask   = (const float*)d_in[3];
  const float* ef     = (const float*)d_in[4];
  const float* We1  = (const float*)d_in[5];
  const float* be1  = (const float*)d_in[6];
  const float* We2  = (const float*)d_in[7];
  const float* be2  = (const float*)d_in[8];
  const float* Watt = (const float*)d_in[9];
  const float* batt = (const float*)d_in[10];
  const float* Wh1  = (const float*)d_in[11];
  const float* bh1  = (const float*)d_in[12];
  const float* Wh2  = (const float*)d_in[13];
  const float* bh2  = (const float*)d_in[14];
  const float* Wx1  = (const float*)d_in[15];
  const float* bx1  = (const float*)d_in[16];
  const float* Wx2  = (const float*)d_in[17];
  const float* bx2  = (const float*)d_in[18];
  const float* Wx3  = (const float*)d_in[19];
  const float* bx3  = (const float*)d_in[20];

  float* ws = (float*)d_ws;
  const size_t RD = (size_t)NROWS * D_MODEL;   // 256K floats = 1MB
  float* HiE = ws + 0 * RD;
  float* HjE = ws + 1 * RD;
  float* HiX = ws + 2 * RD;
  float* HjX = ws + 3 * RD;
  float* NM  = ws + 4 * RD;
  float* H1  = ws + 5 * RD;                    // total 6MB workspace

  float* out_coords = (float*)d_out;           // [8,128,3]
  float* out_feats  = (float*)d_out + NROWS * 3;

  node_proj_kernel<<<512, 256, 0, stream>>>(inv, We1, be1, Wx1, bx1,
                                            HiE, HjE, HiX, HjX);
  egnn_edge_kernel<<<NROWS, 256, 0, stream>>>(
      coords, adj, mask, ef, We1, Watt, batt, We2, be2, Wx1, Wx3, bx3, Wx2,
      bx2, HiE, HjE, HiX, HjX, NM, out_coords);
  h1_kernel<<<128, 256, 0, stream>>>(inv, NM, Wh1, bh1, H1);
  h2_kernel<<<128, 256, 0, stream>>>(H1, Wh2, bh2, mask, out_feats);
}